// Renderer_10350871183719
// MI455X (gfx1250) — compile-verified
//
#include <hip/hip_runtime.h>
#include <hip/hip_bf16.h>

typedef float v2f __attribute__((ext_vector_type(2)));
typedef float v8f __attribute__((ext_vector_type(8)));

#define IS 256
#define NEARP 0.1f
#define FARP 100.0f
// eye z offset = 1/tan(30deg) + 1
#define EYEZ 2.7320508075688772f
// tan(30deg)
#define TANW 0.57735026918962576f

// Workspace layout: one 768B block per 16-face chunk, already in WMMA
// A-matrix lane layout (wave32, f32 16x16x4):
//   block = 3 matrices (w0, w1, inv_z) x 32 lanes x float2
//   lane j    (j<16): (a_j, b_j)   -> A[ M=j ][ K=0..1 ]
//   lane j+16       : (c_j, 0)     -> A[ M=j ][ K=2..3 ]
// so the raster loop loads each A operand with a single coalesced b64 load.
#define CHUNK_FLOATS 192   // 3 * 32 * 2
#define MAT_STRIDE    64   // 32 * 2

// ---------------------------------------------------------------------------
// Kernel 1: per-face setup.
// look_at is identity rotation + translation by EYEZ, then perspective
// divide. Barycentrics and interpolated 1/z are affine in (px,py):
//   w0   = a0 + b0*px + c0*py,  w1 = a1 + b1*px + c1*py,  w2 = 1 - w0 - w1
//   invz = ai + bi*px + ci*py   (coeffs pre-divided by clamped vertex z)
// front-facing & area tests are folded in: inactive faces get a0 = -1e30 so
// w0 < 0 at every pixel and the face never contributes.
// ---------------------------------------------------------------------------
__global__ void face_setup_kernel(const float* __restrict__ verts,
                                  const int* __restrict__ faces,
                                  float* __restrict__ coef,
                                  int NF, int F2pad) {
  int f = blockIdx.x * blockDim.x + threadIdx.x;
  if (f >= F2pad) return;
  int chunk = f >> 4;
  int j     = f & 15;
  float* blk = coef + chunk * CHUNK_FLOATS;

  float a0, b0, c0, a1, b1, c1, ai, bi, ci;
  int F2 = 2 * NF;
  if (f >= F2) {               // padding faces: never hit
    a0 = -1e30f; b0 = 0.f; c0 = 0.f;
    a1 = 0.f; b1 = 0.f; c1 = 0.f;
    ai = 0.f; bi = 0.f; ci = 0.f;
  } else {
    int idx[3];
    if (f < NF) {
      idx[0] = faces[3*f+0]; idx[1] = faces[3*f+1]; idx[2] = faces[3*f+2];
    } else {                   // fill_back: reversed winding
      int g = f - NF;
      idx[0] = faces[3*g+2]; idx[1] = faces[3*g+1]; idx[2] = faces[3*g+0];
    }
    float x[3], y[3], z[3];
    #pragma unroll
    for (int k = 0; k < 3; ++k) {
      float vx = verts[3*idx[k]+0];
      float vy = verts[3*idx[k]+1];
      float vz = verts[3*idx[k]+2];
      float zz = vz + EYEZ;                     // look_at (R == I)
      float zs = (fabsf(zz) < 1e-5f) ? 1e-5f : zz;
      x[k] = vx / (zs * TANW);                  // perspective
      y[k] = vy / (zs * TANW);
      z[k] = zz;
    }
    float area  = (x[1]-x[0])*(y[2]-y[0]) - (x[2]-x[0])*(y[1]-y[0]);
    float areas = (fabsf(area) < 1e-8f) ? 1e-8f : area;
    bool active = (z[0] > NEARP) && (z[1] > NEARP) && (z[2] > NEARP) &&
                  (fabsf(area) > 1e-8f);
    float inv = 1.0f / areas;
    a0 = (x[1]*y[2] - x[2]*y[1]) * inv;
    b0 = (y[1]-y[2]) * inv;
    c0 = (x[2]-x[1]) * inv;
    a1 = (x[2]*y[0] - x[0]*y[2]) * inv;
    b1 = (y[2]-y[0]) * inv;
    c1 = (x[0]-x[2]) * inv;
    float a2 = 1.0f - a0 - a1;
    float b2 = -(b0 + b1);
    float c2 = -(c0 + c1);
    float r0 = 1.0f / fmaxf(z[0], 1e-4f);
    float r1 = 1.0f / fmaxf(z[1], 1e-4f);
    float r2 = 1.0f / fmaxf(z[2], 1e-4f);
    ai = a0*r0 + a1*r1 + a2*r2;
    bi = b0*r0 + b1*r1 + b2*r2;
    ci = c0*r0 + c1*r1 + c2*r2;
    if (!active) { a0 = -1e30f; b0=0.f; c0=0.f; a1=0.f; b1=0.f; c1=0.f;
                   ai=0.f; bi=0.f; ci=0.f; }
  }
  // scatter into WMMA A-matrix lane layout (see header comment)
  v2f* p = (v2f*)blk;
  p[0*32 + j]      = (v2f){a0, b0};
  p[0*32 + j + 16] = (v2f){c0, 0.f};
  p[1*32 + j]      = (v2f){a1, b1};
  p[1*32 + j + 16] = (v2f){c1, 0.f};
  p[2*32 + j]      = (v2f){ai, bi};
  p[2*32 + j + 16] = (v2f){ci, 0.f};
}

// ---------------------------------------------------------------------------
// Kernel 2: WMMA rasterizer. One wave32 owns a 16-pixel strip and loops over
// faces in chunks of 16. Each chunk: three coalesced b64 loads pull the
// pre-swizzled A operands, then 3x v_wmma_f32_16x16x4_f32 evaluate
// w0 / w1 / inv_z for all 16 faces x 16 pixels.
//
// Depth reduction happens in INVERSE-depth space: IEEE 1/x is strictly
// monotone decreasing on positives, so min over faces of zp equals
// 1 / (max over faces of inv_z) bit-exactly. Validity maps to
// 0.01 < inv_z < 10 (== NEAR < zp < FAR); on that region the reference's
// max(inv_z,1e-6) clamp is the identity, so it is elided entirely.
//
// B (4x16): lanes 0-15 hold rows K=0,1 = (1, px); lanes 16-31 hold rows
// K=2,3 = (py, 0) for pixel N = lane&15. D (16x16 f32, 8 VGPRs): lane =
// pixel N, VGPR r = face r (low half) / r+8 (high half).
// ---------------------------------------------------------------------------
__global__ __launch_bounds__(256)
void depth_raster_kernel(const float* __restrict__ coef,
                         float* __restrict__ out, int nchunks) {
  int lane = threadIdx.x & 31;
  int wave = threadIdx.x >> 5;
  int tile = blockIdx.x * (blockDim.x >> 5) + wave;
  int pix  = tile * 16 + (lane & 15);
  float px = (2.0f * (float)(pix & (IS-1)) + 1.0f - (float)IS) * (1.0f/(float)IS);
  float py = (2.0f * (float)(pix >> 8)     + 1.0f - (float)IS) * (1.0f/(float)IS);
  bool hi = lane >= 16;

  v2f B;
  B.x = hi ? py   : 1.0f;   // K=2 : K=0
  B.y = hi ? 0.0f : px;     // K=3 : K=1

  float acc = 0.0f;         // max of inv_z over valid faces; 0 = miss
  for (int ch = 0; ch < nchunks; ++ch) {
    const v2f* cc = (const v2f*)(coef + ch * CHUNK_FLOATS);
    v2f A0 = cc[0*32 + lane];
    v2f A1 = cc[1*32 + lane];
    v2f AI = cc[2*32 + lane];
    if (ch + 1 < nchunks)
      __builtin_prefetch(coef + (ch + 1) * CHUNK_FLOATS + lane * 2, 0, 1);

    v8f zero = {};
    v8f W0 = __builtin_amdgcn_wmma_f32_16x16x4_f32(
        false, A0, false, B, (short)0, zero, false, false);
    v8f W1 = __builtin_amdgcn_wmma_f32_16x16x4_f32(
        false, A1, false, B, (short)0, zero, false, false);
    v8f WI = __builtin_amdgcn_wmma_f32_16x16x4_f32(
        false, AI, false, B, (short)0, zero, false, false);

    #pragma unroll
    for (int r = 0; r < 8; ++r) {
      float w0 = W0[r], w1 = W1[r], wi = WI[r];
      float w2   = 1.0f - w0 - w1;
      float wmin = fminf(fminf(w0, w1), w2);       // v_min3_f32
      // inside & NEAR < zp < FAR, expressed on inv_z (NaN fails all tests,
      // matching the reference's invalid path)
      bool ok = (wmin >= 0.0f) & (wi > 0.01f) & (wi < 10.0f);
      acc = fmaxf(acc, ok ? wi : 0.0f);
    }
  }
  // lanes L and L+16 hold the same pixel (faces r and r+8): combine halves
  acc = fmaxf(acc, __shfl_xor(acc, 16, 32));
  // single exact IEEE divide per pixel reproduces min(zp) bit-exactly
  float depth = (acc > 0.0f) ? (1.0f / acc) : FARP;
  if (lane < 16) out[pix] = depth;
}

extern "C" void kernel_launch(void* const* d_in, const int* in_sizes, int n_in,
                              void* d_out, int out_size, void* d_ws, size_t ws_size,
                              hipStream_t stream) {
  const float* verts = (const float*)d_in[0];   // [1, NV, 3] f32
  const int*   faces = (const int*)d_in[1];     // [1, NF, 3] i32
  float* out = (float*)d_out;                   // [1, 256, 256] f32

  int NF = in_sizes[1] / 3;                     // 5000
  int F2 = 2 * NF;                              // 10000 (fill_back)
  int nchunks = (F2 + 15) / 16;                 // 625
  int F2pad = nchunks * 16;

  float* coef = (float*)d_ws;                   // 768B per chunk (480 KB)

  int sblocks = (F2pad + 255) / 256;
  face_setup_kernel<<<sblocks, 256, 0, stream>>>(verts, faces, coef, NF, F2pad);

  int npix   = out_size;                        // 65536
  int waves  = npix / 16;                       // 4096 waves, 16 px each
  int blocks = waves / 8;                       // 256 threads = 8 waves/block
  depth_raster_kernel<<<blocks, 256, 0, stream>>>(coef, out, nchunks);
}